// GraphLayer_19885698580934
// MI455X (gfx1250) — compile-verified
//
#include <hip/hip_runtime.h>
#include <math.h>

// Problem constants from the reference: N=16384 nodes, M=4 field columns.
#define NN 16384
#define MM 4
#define WAVES_PER_BLOCK 8
#define KSEG (NN / WAVES_PER_BLOCK)   // 2048 K-elements per wave

typedef __attribute__((ext_vector_type(2))) float v2f;
typedef __attribute__((ext_vector_type(8))) float v8f;

// out = b + alpha * d^gamma * z + beta * d^(gamma-1) * (A @ z)
// One workgroup = one 16-row tile of A@z. 8 waves split the K dimension,
// each accumulating via V_WMMA_F32_16X16X4_F32 (exact fp32, memory-bound
// on the 1 GiB read of A). LDS reduction + pointwise epilogue at the end.
__global__ __launch_bounds__(32 * WAVES_PER_BLOCK)
void graph_layer_wmma(const float* __restrict__ z,
                      const float* __restrict__ params,
                      const float* __restrict__ A,
                      const float* __restrict__ Dm,
                      float* __restrict__ out) {
    __shared__ float red[WAVES_PER_BLOCK * 16 * MM];   // 2 KB

    const int tid  = threadIdx.x;
    const int lane = tid & 31;
    const int wave = tid >> 5;
    const int half = lane >> 4;      // 0: K lo pair, 1: K hi pair
    const int n    = lane & 15;      // A: row-in-tile (M); B/C: column (N)
    const int r0   = blockIdx.x << 4;
    const int row  = r0 + n;

    const int  kbase  = wave * KSEG;
    const int  koff   = half * 2;
    const bool nvalid = (n < MM);    // only 4 of the 16 N-columns are real

    // Per-lane A stream: A[row][kbase+koff ...], 8 bytes per WMMA step.
    const float* ap = A + (size_t)row * NN + (size_t)(kbase + koff);
    // Per-lane z stream: z[kbase+koff + step][n&3]; zeroed later if n>=4.
    const float* zp = z + (size_t)(kbase + koff) * MM + (n & 3);

    v8f c = {};
    #pragma unroll 4
    for (int k = 0; k < KSEG; k += 4) {
        // A-tile operand: lane<16 -> K = k,k+1 ; lane>=16 -> K = k+2,k+3
        v2f av = *reinterpret_cast<const v2f*>(ap);
        // B-tile operand (4x16): VGPR v holds K = k + 2*half + v, col = n
        float zb0 = zp[0];
        float zb1 = zp[MM];
        v2f bv;
        bv.x = nvalid ? zb0 : 0.0f;  // cndmask, no divergence: EXEC stays all-1s
        bv.y = nvalid ? zb1 : 0.0f;
        // D = A_tile(16x4) x B_tile(4x16) + C   (exact fp32)
        c = __builtin_amdgcn_wmma_f32_16x16x4_f32(
                /*neg_a=*/false, av, /*neg_b=*/false, bv,
                /*c_mod=*/(short)0, c, /*reuse_a=*/false, /*reuse_b=*/false);
        ap += 4;
        zp += 4 * MM;
    }

    // Dump this wave's 16x4 useful slice of C to LDS.
    // C layout: VGPR v, lane half -> M = v + 8*half; column N = n.
    if (nvalid) {
        #pragma unroll
        for (int v = 0; v < 8; ++v) {
            const int m = v + half * 8;
            red[(wave * 16 + m) * MM + n] = c[v];
        }
    }
    __syncthreads();

    // 64 outputs per tile: thread j -> (m = j/4, col = j%4).
    if (tid < 16 * MM) {
        float acc = 0.0f;
        #pragma unroll
        for (int w = 0; w < WAVES_PER_BLOCK; ++w)
            acc += red[w * 16 * MM + tid];

        const int m  = tid >> 2;
        const int j  = tid & 3;
        const int r  = r0 + m;

        // param transforms (exactly as reference)
        const float alpha = expf(params[0]);
        const float beta  = -alpha * tanhf(params[1]);
        const float gamma = expf(params[2]);
        const float bias  = params[3];

        const float d    = Dm[(size_t)r * NN + r];   // diagonal entry, d > 0
        const float logd = logf(d);
        const float dg   = expf(gamma * logd);          // d^gamma
        const float dgm1 = expf((gamma - 1.0f) * logd); // d^(gamma-1)

        out[(size_t)r * MM + j] =
            bias + alpha * dg * z[(size_t)r * MM + j] + beta * dgm1 * acc;
    }
}

extern "C" void kernel_launch(void* const* d_in, const int* in_sizes, int n_in,
                              void* d_out, int out_size, void* d_ws, size_t ws_size,
                              hipStream_t stream) {
    (void)in_sizes; (void)n_in; (void)out_size; (void)d_ws; (void)ws_size;
    const float* z      = (const float*)d_in[0];
    const float* params = (const float*)d_in[1];
    const float* A      = (const float*)d_in[2];
    const float* Dm     = (const float*)d_in[3];
    float*       out    = (float*)d_out;

    dim3 grid(NN / 16);                 // 1024 row-tiles
    dim3 block(32 * WAVES_PER_BLOCK);   // 8 waves (wave32)
    hipLaunchKernelGGL(graph_layer_wmma, grid, block, 0, stream,
                       z, params, A, Dm, out);
}